// StateAwareGraphRouting_79525614453051
// MI455X (gfx1250) — compile-verified
//
#include <hip/hip_runtime.h>

#define NN 8192
#define KK 64
#define DKK 64
#define GG 16

typedef __attribute__((ext_vector_type(2))) float v2f;
typedef __attribute__((ext_vector_type(8))) float v8f;

// ---- workspace float offsets ----
#define PART_P 0          // [64][64] partial colsums of P_in
#define PART_A 4096       // [64][64] partial colsums of A_in
#define PARTSQ 8192       // [64][2]  partial sq-sums (P, A)
#define COLM   8320       // m_in_P[64], m_in_A[64]
#define M2     8448       // m2P, m2A
#define KG     8512       // k_g [16][64]
#define PROP   9536       // propT [64][8192]  (transposed propagated)

static __device__ inline v8f wmma4(v2f a, v2f b, v8f c) {
    return __builtin_amdgcn_wmma_f32_16x16x4_f32(
        /*neg_a=*/false, a, /*neg_b=*/false, b,
        /*c_mod=*/(short)0, c, /*reuse_a=*/false, /*reuse_b=*/false);
}

static __device__ inline unsigned rotl32(unsigned x, int r) {
    return (x << r) | (x >> (32 - r));
}

// JAX Threefry2x32 with key(42) -> (k0,k1) = (0,42)
static __device__ inline void threefry2x32_42(unsigned c0, unsigned c1,
                                              unsigned& o0, unsigned& o1) {
    const unsigned k0 = 0u, k1 = 42u;
    const unsigned k2 = k0 ^ k1 ^ 0x1BD11BDAu;
    unsigned x0 = c0 + k0, x1 = c1 + k1;
#define TF_R(r) { x0 += x1; x1 = rotl32(x1, (r)); x1 ^= x0; }
    TF_R(13) TF_R(15) TF_R(26) TF_R(6)   x0 += k1; x1 += k2 + 1u;
    TF_R(17) TF_R(29) TF_R(16) TF_R(24)  x0 += k2; x1 += k0 + 2u;
    TF_R(13) TF_R(15) TF_R(26) TF_R(6)   x0 += k0; x1 += k1 + 3u;
    TF_R(17) TF_R(29) TF_R(16) TF_R(24)  x0 += k1; x1 += k2 + 4u;
    TF_R(13) TF_R(15) TF_R(26) TF_R(6)   x0 += k2; x1 += k0 + 5u;
#undef TF_R
    o0 = x0; o1 = x1;
}

// ---------------- Kernel 1: partial column sums / sq sums ----------------
__global__ __launch_bounds__(256) void k_colreduce(const float* __restrict__ Pin,
                                                   const float* __restrict__ Ain,
                                                   float* __restrict__ ws) {
    __shared__ float red[256];
    const int b = blockIdx.x, tid = threadIdx.x;
    const int col = tid & 63, rb = tid >> 6;
    float sP = 0.f, sA = 0.f, qP = 0.f, qA = 0.f;
    for (int r = rb; r < 128; r += 4) {
        const int row = b * 128 + r;
        float v = Pin[row * 64 + col]; sP += v; qP += v * v;
        float w = Ain[row * 64 + col]; sA += w; qA += w * w;
    }
    red[tid] = sP; __syncthreads();
    if (tid < 64) ws[PART_P + b * 64 + tid] = red[tid] + red[tid + 64] + red[tid + 128] + red[tid + 192];
    __syncthreads();
    red[tid] = sA; __syncthreads();
    if (tid < 64) ws[PART_A + b * 64 + tid] = red[tid] + red[tid + 64] + red[tid + 128] + red[tid + 192];
    __syncthreads();
    red[tid] = qP; __syncthreads();
    for (int s = 128; s > 0; s >>= 1) { if (tid < s) red[tid] += red[tid + s]; __syncthreads(); }
    if (tid == 0) ws[PARTSQ + b * 2 + 0] = red[0];
    __syncthreads();
    red[tid] = qA; __syncthreads();
    for (int s = 128; s > 0; s >>= 1) { if (tid < s) red[tid] += red[tid + s]; __syncthreads(); }
    if (tid == 0) ws[PARTSQ + b * 2 + 1] = red[0];
}

// ---------------- Kernel 1b: finalize means ----------------
__global__ __launch_bounds__(64) void k_colfinal(float* __restrict__ ws) {
    const int t = threadIdx.x;
    float sP = 0.f, sA = 0.f;
    for (int b = 0; b < 64; b++) { sP += ws[PART_P + b * 64 + t]; sA += ws[PART_A + b * 64 + t]; }
    ws[COLM + t]      = sP / (float)NN;
    ws[COLM + 64 + t] = sA / (float)NN;
    if (t == 0) {
        float m2P = 0.f, m2A = 0.f;
        for (int b = 0; b < 64; b++) { m2P += ws[PARTSQ + b * 2]; m2A += ws[PARTSQ + b * 2 + 1]; }
        ws[M2]     = m2P / (float)(NN * KK);
        ws[M2 + 1] = m2A / (float)(NN * KK);
    }
}

// ---------------- k_g = W_s.reshape(G,-1) @ W_k_w.T + b ----------------
__global__ __launch_bounds__(64) void k_kg(const float* __restrict__ Ws,
                                           const float* __restrict__ Wkw,
                                           const float* __restrict__ Wkb,
                                           float* __restrict__ ws) {
    const int g = blockIdx.x, d = threadIdx.x;
    const float* a = Ws + g * 4096;
    const float* w = Wkw + d * 4096;
    float acc = 0.f;
#pragma unroll 4
    for (int m = 0; m < 4096; m++) acc += a[m] * w[m];
    ws[KG + g * 64 + d] = acc + Wkb[d];
}

// ---------------- Kernel 2: routing (1 wave per 16 nodes) ----------------
__global__ __launch_bounds__(32) void k_route(
    const float* __restrict__ z, const float* __restrict__ Pref, const float* __restrict__ Aref,
    const float* __restrict__ Wi_w, const float* __restrict__ Wi_b,
    const float* __restrict__ Wg_w, const float* __restrict__ Wg_b,
    const float* __restrict__ Wsys_w, const float* __restrict__ Wsys_b,
    const float* __restrict__ Ws, float* __restrict__ ws, float* __restrict__ out_s) {
    __shared__ float q_lds[16 * 64];
    __shared__ float d_lds[2 * 16];   // [0][m]=dA, [1][m]=dP
    __shared__ float a_lds[16 * 16];  // alpha tile

    const int lane = threadIdx.x;
    const int m0   = blockIdx.x * 16;
    const int l15  = lane & 15;
    const int hi   = lane >> 4;       // 0 or 1
    const int koff = hi * 2;

    // ---- per-node deltas: lanes 0-15 -> dP, lanes 16-31 -> dA ----
    {
        const float* Xref = (hi == 0) ? Pref : Aref;
        const float* cm   = ws + COLM + hi * 64;
        const float  m2   = ws[M2 + hi];
        const float* row  = Xref + (m0 + l15) * 64;
        float dot = 0.f, sq = 0.f;
        for (int k = 0; k < 64; k++) { float v = row[k]; dot += v * cm[k]; sq += v * v; }
        float dval = m2 - 2.0f * dot * (1.0f / 64.0f) + sq * (1.0f / 64.0f);
        d_lds[(hi == 0 ? 1 : 0) * 16 + l15] = dval;
    }
    __syncthreads();

    // ---- z tile in WMMA A-layout ----
    v2f za[16];
    {
        const float* zr = z + (m0 + l15) * 64 + koff;
#pragma unroll
        for (int s = 0; s < 16; s++) za[s] = *(const v2f*)(zr + s * 4);
    }

    // ---- gate = z@Wg^T, ind = z@Wi^T ----
    v8f gacc[4], iacc[4];
#pragma unroll
    for (int t = 0; t < 4; t++) {
        v8f gA = {}; v8f iA = {};
        const int d = t * 16 + l15;
#pragma unroll
        for (int s = 0; s < 16; s++) {
            v2f bg = *(const v2f*)(Wg_w + d * 64 + s * 4 + koff);
            v2f bi = *(const v2f*)(Wi_w + d * 64 + s * 4 + koff);
            gA = wmma4(za[s], bg, gA);
            iA = wmma4(za[s], bi, iA);
        }
        gacc[t] = gA; iacc[t] = iA;
    }

    // ---- q = ind + b_ind + sigmoid(gate + b_gate) * sys_feat ; spill to LDS ----
#pragma unroll
    for (int t = 0; t < 4; t++) {
        const int d = t * 16 + l15;
        const float bg = Wg_b[d], bi = Wi_b[d], bs = Wsys_b[d];
        const float w0 = Wsys_w[d * 2 + 0], w1 = Wsys_w[d * 2 + 1];
#pragma unroll
        for (int r = 0; r < 8; r++) {
            const int M = r + hi * 8;
            const float dA = d_lds[0 * 16 + M], dP = d_lds[1 * 16 + M];
            const float gate = 1.0f / (1.0f + __expf(-(gacc[t][r] + bg)));
            const float sys  = dA * w0 + dP * w1 + bs;
            q_lds[M * 64 + d] = iacc[t][r] + bi + gate * sys;
        }
    }
    __syncthreads();

    // ---- s_ng = (q @ k_g^T) / sqrt(DK) ----
    v8f sacc = {};
#pragma unroll
    for (int s = 0; s < 16; s++) {
        v2f qa = *(const v2f*)(&q_lds[l15 * 64 + s * 4 + koff]);
        v2f kb = *(const v2f*)(ws + KG + l15 * 64 + s * 4 + koff);
        sacc = wmma4(qa, kb, sacc);
    }

    // ---- store s_ng, Gumbel (exact JAX Threefry2x32), softmax -> alpha ----
    float sv[8], av[8];
#pragma unroll
    for (int r = 0; r < 8; r++) {
        const int M = r + hi * 8;
        const int n = m0 + M;
        const float s_out = sacc[r] * 0.125f;  // 1/sqrt(64)
        out_s[n * 16 + l15] = s_out;
        const unsigned idx = (unsigned)(n * 16 + l15);
        const unsigned i0 = (idx < 65536u) ? idx : (idx - 65536u);
        unsigned r0, r1; threefry2x32_42(i0, i0 + 65536u, r0, r1);
        const unsigned bits = (idx < 65536u) ? r0 : r1;
        float u = __uint_as_float((bits >> 9) | 0x3F800000u) - 1.0f;
        u = fmaxf(u, 0.0f);
        const float gum = -__logf(-__logf(u + 1e-10f) + 1e-10f);
        sv[r] = s_out + gum;  // TAU = 1
    }
#pragma unroll
    for (int r = 0; r < 8; r++) {
        float x = sv[r];
        float mx = x;
#pragma unroll
        for (int m = 8; m >= 1; m >>= 1) mx = fmaxf(mx, __shfl_xor(mx, m, 32));
        float e = __expf(x - mx);
        float ssum = e;
#pragma unroll
        for (int m = 8; m >= 1; m >>= 1) ssum += __shfl_xor(ssum, m, 32);
        av[r] = e / ssum;
    }
#pragma unroll
    for (int r = 0; r < 8; r++) a_lds[(r + hi * 8) * 16 + l15] = av[r];
    __syncthreads();

    // ---- propagated[n] = sum_g alpha[n,g] * (z @ W_s[g]^T)[n] ----
    v8f pacc[4] = {};
    for (int g = 0; g < GG; g++) {
        float al[8];
#pragma unroll
        for (int r = 0; r < 8; r++) al[r] = a_lds[(r + hi * 8) * 16 + g];
        const float* Wg = Ws + g * 4096;
#pragma unroll
        for (int t = 0; t < 4; t++) {
            v8f tacc = {};
            const int i = t * 16 + l15;
#pragma unroll
            for (int s = 0; s < 16; s++) {
                v2f b = *(const v2f*)(Wg + i * 64 + s * 4 + koff);
                tacc = wmma4(za[s], b, tacc);
            }
#pragma unroll
            for (int r = 0; r < 8; r++) pacc[t][r] += al[r] * tacc[r];
        }
    }
    // store TRANSPOSED: propT[col][n]  -> enables b64 B-operand loads in k_bigmm
    float* propT = ws + PROP;
#pragma unroll
    for (int t = 0; t < 4; t++)
#pragma unroll
        for (int r = 0; r < 8; r++) {
            const int M = r + hi * 8;
            propT[(t * 16 + l15) * NN + (m0 + M)] = pacc[t][r];
        }
}

// ---------------- Kernel 3: h = leaky_relu(R @ propagated) ----------------
// propT is [64][8192] (column-major propagated): B-operand pair for WMMA step k
// (rows k+koff, k+koff+1 of B at column c) is contiguous -> single b64 load.
__global__ __launch_bounds__(256) void k_bigmm(const float* __restrict__ R,
                                               const float* __restrict__ propT,
                                               float* __restrict__ out_h) {
    __shared__ float part[8][1024];
    const int tid  = threadIdx.x;
    const int lane = tid & 31;
    const int wave = tid >> 5;
    const int m0   = blockIdx.x * 16;
    const int l15  = lane & 15;
    const int koff = (lane >> 4) * 2;

    v8f acc[4] = {};
    const int k0 = wave * 1024;  // 8 waves split K = 8192
    const float* arow = R + (size_t)(m0 + l15) * NN + koff;
    const float* b0 = propT + (0 * 16 + l15) * NN + koff;
    const float* b1 = propT + (1 * 16 + l15) * NN + koff;
    const float* b2 = propT + (2 * 16 + l15) * NN + koff;
    const float* b3 = propT + (3 * 16 + l15) * NN + koff;
#pragma unroll 2
    for (int k = k0; k < k0 + 1024; k += 4) {
        v2f a = *(const v2f*)(arow + k);
        acc[0] = wmma4(a, *(const v2f*)(b0 + k), acc[0]);
        acc[1] = wmma4(a, *(const v2f*)(b1 + k), acc[1]);
        acc[2] = wmma4(a, *(const v2f*)(b2 + k), acc[2]);
        acc[3] = wmma4(a, *(const v2f*)(b3 + k), acc[3]);
    }
#pragma unroll
    for (int t = 0; t < 4; t++)
#pragma unroll
        for (int r = 0; r < 8; r++) {
            const int M = r + (lane >> 4) * 8;
            part[wave][M * 64 + t * 16 + l15] = acc[t][r];
        }
    __syncthreads();
#pragma unroll
    for (int i = tid * 4; i < tid * 4 + 4; i++) {
        float s = 0.f;
#pragma unroll
        for (int w = 0; w < 8; w++) s += part[w][i];
        out_h[m0 * 64 + i] = (s >= 0.0f) ? s : 0.01f * s;
    }
}

extern "C" void kernel_launch(void* const* d_in, const int* in_sizes, int n_in,
                              void* d_out, int out_size, void* d_ws, size_t ws_size,
                              hipStream_t stream) {
    const float* z      = (const float*)d_in[0];
    const float* Pref   = (const float*)d_in[1];
    const float* Pin    = (const float*)d_in[2];
    const float* Aref   = (const float*)d_in[3];
    const float* Ain    = (const float*)d_in[4];
    const float* R      = (const float*)d_in[5];
    const float* Wi_w   = (const float*)d_in[6];
    const float* Wi_b   = (const float*)d_in[7];
    const float* Wg_w   = (const float*)d_in[8];
    const float* Wg_b   = (const float*)d_in[9];
    const float* Wsys_w = (const float*)d_in[10];
    const float* Wsys_b = (const float*)d_in[11];
    const float* Ws     = (const float*)d_in[12];
    const float* Wk_w   = (const float*)d_in[13];
    const float* Wk_b   = (const float*)d_in[14];
    float* out = (float*)d_out;
    float* ws  = (float*)d_ws;

    k_colreduce<<<64, 256, 0, stream>>>(Pin, Ain, ws);
    k_colfinal<<<1, 64, 0, stream>>>(ws);
    k_kg<<<16, 64, 0, stream>>>(Ws, Wk_w, Wk_b, ws);
    k_route<<<512, 32, 0, stream>>>(z, Pref, Aref, Wi_w, Wi_b, Wg_w, Wg_b,
                                    Wsys_w, Wsys_b, Ws, ws, out + NN * DKK);
    k_bigmm<<<512, 256, 0, stream>>>(R, ws + PROP, out);
}